// Model_21474836480302
// MI455X (gfx1250) — compile-verified
//
#include <hip/hip_runtime.h>

// ---------------------------------------------------------------------------
// ACE-style descriptor forward for MI455X (gfx1250, wave32, WMMA).
//
// Stage 1: zero A_s workspace [N,4,16,8] (61.4 MB -> L2 resident) and d_out.
// Stage 2: edge kernel: Y(16) x B(8) outer product, 128 global_atomic_add_f32
//          into A_s[dst*4 + species[src]].
// Stage 3: atom kernel: one wave32 per atom; contraction over species
//          (K=4) done with V_WMMA_F32_16X16X4_F32 (8 tiles of 16 am-rows x
//          16 channels), then nu1/nu2 nonlinearity, weighted reduce,
//          atomic add into structure energy.
// ---------------------------------------------------------------------------

typedef __attribute__((ext_vector_type(2))) float v2f;
typedef __attribute__((ext_vector_type(8))) float v8f;

#define RCUT  5.0f
#define PI_F  3.14159265358979323846f

// ---------------------------------------------------------------- stage 1 ---
__global__ void zero_ws_kernel(float4* __restrict__ ws4, long n4,
                               float* __restrict__ out, int out_n) {
  long i = (long)blockIdx.x * blockDim.x + threadIdx.x;
  long stride = (long)gridDim.x * blockDim.x;
  float4 z = make_float4(0.f, 0.f, 0.f, 0.f);
  for (long k = i; k < n4; k += stride) ws4[k] = z;
  if (i < out_n) out[i] = 0.f;
}

// ---------------------------------------------------------------- stage 2 ---
__global__ void edge_kernel(const float* __restrict__ pos,
                            const int* __restrict__ species,
                            const int* __restrict__ pairs,
                            float* __restrict__ As, int E) {
  int e = blockIdx.x * blockDim.x + threadIdx.x;
  if (e >= E) return;
  int src = pairs[2 * e + 0];
  int dst = pairs[2 * e + 1];

  float dx = pos[3 * src + 0] - pos[3 * dst + 0];
  float dy = pos[3 * src + 1] - pos[3 * dst + 1];
  float dz = pos[3 * src + 2] - pos[3 * dst + 2];
  float d  = sqrtf(dx * dx + dy * dy + dz * dz + 1e-12f);
  float inv = 1.0f / d;
  float x = dx * inv, y = dy * inv, z = dz * inv;
  float x2 = x * x, y2 = y * y, z2 = z * z;

  // real spherical harmonics up to l=3
  float Y[16];
  Y[0]  = 0.28209479177387814f;
  Y[1]  = 0.4886025119029199f * y;
  Y[2]  = 0.4886025119029199f * z;
  Y[3]  = 0.4886025119029199f * x;
  Y[4]  = 1.0925484305920792f * x * y;
  Y[5]  = 1.0925484305920792f * y * z;
  Y[6]  = 0.31539156525252005f * (3.0f * z2 - 1.0f);
  Y[7]  = 1.0925484305920792f * x * z;
  Y[8]  = 0.5462742152960396f * (x2 - y2);
  Y[9]  = 0.5900435899266435f * y * (3.0f * x2 - y2);
  Y[10] = 2.890611442640554f * x * y * z;
  Y[11] = 0.4570457994644658f * y * (5.0f * z2 - 1.0f);
  Y[12] = 0.3731763325901154f * z * (5.0f * z2 - 3.0f);
  Y[13] = 0.4570457994644658f * x * (5.0f * z2 - 1.0f);
  Y[14] = 1.445305721320277f * z * (x2 - y2);
  Y[15] = 0.5900435899266435f * x * (x2 - 3.0f * y2);

  // radial basis with cosine cutoff
  float fc = (d < RCUT) ? 0.5f * (cosf(PI_F * d / RCUT) + 1.0f) : 0.0f;
  float s  = 0.6324555320336759f * fc * inv;  // sqrt(2/RC) * fc / d
  float ang = (PI_F / RCUT) * d;
  float B[8];
#pragma unroll
  for (int m = 0; m < 8; ++m) B[m] = s * sinf((float)(m + 1) * ang);

  int seg = dst * 4 + species[src];
  float* base = As + (size_t)seg * 128;
#pragma unroll
  for (int a = 0; a < 16; ++a) {
    float ya = Y[a];
#pragma unroll
    for (int m = 0; m < 8; ++m) {
      unsafeAtomicAdd(base + a * 8 + m, ya * B[m]);  // global_atomic_add_f32
    }
  }
}

// ---------------------------------------------------------------- stage 3 ---
// One wave32 per atom. A_s[atom] is a 4(species) x 128(am) block.
// WMMA tile t: rows am = 16t + M, M = (lane&15) + 8*(lane>=16)? per ISA layout:
//   A operand 16x4 f32: lanes 0-15 M=lane, VGPR0 K=0/VGPR1 K=1;
//                       lanes 16-31 M=lane-16, VGPR0 K=2/VGPR1 K=3.
//   B operand 4x16 f32: VGPR0 row K=0 (lo lanes) / K=2 (hi lanes), VGPR1 K=1/K=3.
//   C/D 16x16 f32: VGPR j -> M=j (lo lanes), M=j+8 (hi lanes), N=lane&15.
// With am = 16t + j + 8*hi:  m = j  and  a = 2t + hi.
__global__ void atom_kernel(const float* __restrict__ As,
                            const int* __restrict__ species,
                            const int* __restrict__ sid,
                            const float* __restrict__ emb,     // [4,16]
                            const float* __restrict__ wout,    // [128]
                            const float* __restrict__ compw,   // [4]
                            float* __restrict__ out, int N) {
  int atom = blockIdx.x * 4 + (threadIdx.x >> 5);
  if (atom >= N) return;
  int lane = threadIdx.x & 31;
  int hi = lane >> 4;
  int c  = lane & 15;

  int spec = species[atom];
  float emb_c = emb[spec * 16 + c];

  float wv[8];
#pragma unroll
  for (int j = 0; j < 8; ++j) wv[j] = wout[j * 16 + c];

  // B operand: embeddings, rows = species (K), cols = channel (N)
  v2f b;
  b.x = emb[(2 * hi + 0) * 16 + c];
  b.y = emb[(2 * hi + 1) * 16 + c];

  const float* Ab = As + (size_t)atom * 512;  // 4 species * 128 am

  v8f acc[8];
#pragma unroll
  for (int t = 0; t < 8; ++t) {
    v2f a;
    a.x = Ab[(2 * hi + 0) * 128 + 16 * t + c];
    a.y = Ab[(2 * hi + 1) * 128 + 16 * t + c];
    v8f cz = {0.f, 0.f, 0.f, 0.f, 0.f, 0.f, 0.f, 0.f};
    acc[t] = __builtin_amdgcn_wmma_f32_16x16x4_f32(
        /*neg_a=*/false, a, /*neg_b=*/false, b,
        /*c_mod=*/(short)0, cz, /*reuse_a=*/false, /*reuse_b=*/false);
  }

  // nu1 + nu2 nonlinearity and weighted energy reduction.
  float e = 0.f;
#pragma unroll
  for (int t = 0; t < 8; ++t) {
    int aidx = 2 * t + hi;
    float invs = (aidx == 0) ? 1.0f
               : (aidx < 4)  ? 0.5773502691896258f   // 1/sqrt(3)
               : (aidx < 9)  ? 0.4472135954999579f   // 1/sqrt(5)
                             : 0.3779644730092272f;  // 1/sqrt(7)
#pragma unroll
    for (int j = 0; j < 8; ++j) {
      float v = acc[t][j];
      if (aidx == 0) v += emb_c;          // init term added to A[:,0,:]
      float contrib = v * v * invs;       // nu2 partial
      if (aidx == 0) contrib += v;        // nu1
      e += contrib * emb_c * wv[j];       // hidden * w_out
    }
  }

  // wave32 butterfly reduction (DS permute hardware)
#pragma unroll
  for (int off = 16; off > 0; off >>= 1) e += __shfl_xor(e, off, 32);

  if (lane == 0) {
    unsafeAtomicAdd(&out[sid[atom]], e + compw[spec]);
  }
}

// ---------------------------------------------------------------------------
extern "C" void kernel_launch(void* const* d_in, const int* in_sizes, int n_in,
                              void* d_out, int out_size, void* d_ws, size_t ws_size,
                              hipStream_t stream) {
  const float* pos     = (const float*)d_in[0];
  const int*   species = (const int*)d_in[1];
  const int*   pairs   = (const int*)d_in[2];
  const int*   sid     = (const int*)d_in[3];
  const float* emb     = (const float*)d_in[4];
  const float* wout    = (const float*)d_in[5];
  const float* compw   = (const float*)d_in[6];
  float*       out     = (float*)d_out;
  float*       As      = (float*)d_ws;   // [N,4,16,8] = N*512 floats

  int N = in_sizes[0] / 3;
  int E = in_sizes[2] / 2;

  // Stage 1: zero A_s and output
  long n4 = (long)N * 512 / 4;
  zero_ws_kernel<<<2048, 256, 0, stream>>>((float4*)As, n4, out, out_size);

  // Stage 2: edge scatter
  edge_kernel<<<(E + 255) / 256, 256, 0, stream>>>(pos, species, pairs, As, E);

  // Stage 3: per-atom WMMA contraction + energy
  atom_kernel<<<(N + 3) / 4, 128, 0, stream>>>(As, species, sid, emb, wout,
                                               compw, out, N);
}